// MultiScaleDETR_43997644981053
// MI455X (gfx1250) — compile-verified
//
#include <hip/hip_runtime.h>
#include <math.h>

typedef __attribute__((ext_vector_type(2))) float v2f;
typedef __attribute__((ext_vector_type(4))) float v4f;
typedef __attribute__((ext_vector_type(8))) float v8f;

#define NH   8
#define NL   3
#define NP   4
#define DIM  256
#define HD   32
#define BATCH 4
#define QN   8400
#define BQ   (BATCH * QN)   // 33600
#define STOT 8400           // 6400 + 1600 + 400

// ---------------------------------------------------------------------------
// Tiled GEMM on the f32 WMMA pipe:  C[z] = A[z] * B + bias, optional
// tanh(x)*0.5 epilogue (ACT).  Block = 256 threads = 8 waves.
// Block tile 128x64; each wave owns a 16x64 slab (4 accumulators), so each
// A fragment feeds 4 back-to-back V_WMMA_F32_16X16X4_F32 ops.
// K staged through LDS in 16-deep slabs.  ACOL=1 reads A column-major
// (addr = k*lda + m): NCHW feature maps [D][S] are [S][D] col-major, ld = S.
// Requires M, N, lda multiples of 4 (true for all calls here).
//
// LDS layouts:
//   As: stride 20 -> rows 16B-aligned (b128 stores, b64 frag loads);
//       frag-read banks: half0 hits =0 mod 4, half1 hits =2 mod 4 -> no conflict.
//   Bs: K-pair interleaved, element (k,n) -> Bs[(k>>1)*160 + n*2 + (k&1)];
//       a B-fragment (rows K,K+1) is one aligned 8B word -> ds_load_b64;
//       pair stride 160 ≡ 32 banks -> halves use complementary bank windows.
// Staging is branch-free: pointers hoisted & strided per slab, OOB handled by
// clamped addresses + cndmask zero-fill, 128-bit global loads.
// ---------------------------------------------------------------------------
template<int ACOL, int ACT>
__global__ __launch_bounds__(256)
void gemm_wmma_f32(const float* __restrict__ A, const float* __restrict__ Bm,
                   const float* __restrict__ bias, float* __restrict__ C,
                   int M, int N, int K, int lda,
                   long a_bstride, long c_bstride)
{
    __shared__ float As[128 * 20];   // 10 KB
    __shared__ float Bs[8 * 160];    // 5 KB (8 K-pairs)
    const int tid  = threadIdx.x;
    const int lane = tid & 31;
    const int wid  = tid >> 5;       // 8 waves along M (16 rows each)
    const int half = lane >> 4;      // lane group per ISA 7.12.2
    const int l16  = lane & 15;
    const int bm   = blockIdx.y * 128;
    const int bn   = blockIdx.x * 64;
    const long abase = (long)blockIdx.z * a_bstride;
    const long cbase = (long)blockIdx.z * c_bstride;

    // ---- hoisted staging descriptors (2 A-quads + 1 B-quad per thread) ----
    const float* apx[2];
    int  a_sto[2];
    bool a_ok[2];
    #pragma unroll
    for (int it = 0; it < 2; ++it) {
        int qi = tid + it * 256;               // quad index (512 A-quads)
        if (ACOL) {
            int k  = qi >> 5;                  // 16 K-rows x 32 m-quads
            int m4 = (qi & 31) * 4;
            int gm = bm + m4;
            a_ok[it]  = gm < M;
            apx[it]   = A + abase + (long)k * lda + (a_ok[it] ? gm : 0);
            a_sto[it] = m4 * 20 + k;
        } else {
            int m  = qi >> 2;                  // 128 rows x 4 k-quads
            int k4 = (qi & 3) * 4;
            int gm = bm + m;
            a_ok[it]  = gm < M;
            apx[it]   = A + abase + (long)(a_ok[it] ? gm : 0) * lda + k4;
            a_sto[it] = m * 20 + k4;
        }
    }
    const long astep = ACOL ? (long)16 * lda : 16;

    const float* bpx;
    int  b_sto;
    bool b_ok;
    {
        int k  = tid >> 4;                     // 16 K-rows x 16 n-quads
        int n4 = (tid & 15) * 4;
        int gn = bn + n4;
        b_ok  = gn < N;
        bpx   = Bm + (long)k * N + (b_ok ? gn : 0);
        b_sto = (k >> 1) * 160 + n4 * 2 + (k & 1);
    }
    const long bstep = (long)16 * N;

    v8f acc[4];
    #pragma unroll
    for (int j = 0; j < 4; ++j)
        acc[j] = (v8f){0.f,0.f,0.f,0.f,0.f,0.f,0.f,0.f};

    for (int k0 = 0; k0 < K; k0 += 16) {
        // stage A 128x16
        #pragma unroll
        for (int it = 0; it < 2; ++it) {
            v4f q = *(const v4f*)apx[it];
            if (!a_ok[it]) q = (v4f){0.f, 0.f, 0.f, 0.f};
            if (ACOL) {                        // quad spans 4 m-rows
                As[a_sto[it] +  0] = q.x;
                As[a_sto[it] + 20] = q.y;
                As[a_sto[it] + 40] = q.z;
                As[a_sto[it] + 60] = q.w;
            } else {                           // quad is contiguous in a row
                *(v4f*)&As[a_sto[it]] = q;
            }
            apx[it] += astep;
        }
        // stage B 16x64 (pair-interleaved)
        {
            v4f q = *(const v4f*)bpx;
            if (!b_ok) q = (v4f){0.f, 0.f, 0.f, 0.f};
            Bs[b_sto + 0] = q.x;
            Bs[b_sto + 2] = q.y;
            Bs[b_sto + 4] = q.z;
            Bs[b_sto + 6] = q.w;
            bpx += bstep;
        }
        __syncthreads();

        #pragma unroll
        for (int kk = 0; kk < 16; kk += 4) {
            // A frag 16x4: lanes 0-15 -> K=kk,kk+1 ; lanes 16-31 -> K=kk+2,kk+3
            v2f a = *(const v2f*)(As + (wid * 16 + l16) * 20 + kk + 2 * half);
            // B frags: K-pair (kk/2 + half), one 8B word per lane per j-tile
            const float* bp = Bs + ((kk >> 1) + half) * 160 + l16 * 2;
            #pragma unroll
            for (int j = 0; j < 4; ++j) {
                v2f b = *(const v2f*)(bp + j * 32);
                acc[j] = __builtin_amdgcn_wmma_f32_16x16x4_f32(
                    false, a, false, b, (short)0, acc[j], false, false);
            }
        }
        __syncthreads();
    }

    // C/D layout: VGPR r -> M = r (lanes 0-15) / r+8 (lanes 16-31), N = l16
    #pragma unroll
    for (int r = 0; r < 8; ++r) {
        int row = bm + wid * 16 + r + 8 * half;
        if (row < M) {
            #pragma unroll
            for (int j = 0; j < 4; ++j) {
                int col = bn + j * 16 + l16;
                if (col < N) {
                    float v = acc[j][r] + bias[col];
                    if (ACT) v = tanhf(v) * 0.5f;
                    C[cbase + (long)row * N + col] = v;
                }
            }
        }
    }
}

// ---------------------------------------------------------------------------
// Softmax over the 12 (level,point) logits per (b,q,head), in place.
// ---------------------------------------------------------------------------
__global__ __launch_bounds__(256)
void softmax12(float* __restrict__ aw, int n)
{
    int i = blockIdx.x * blockDim.x + threadIdx.x;
    if (i >= n) return;
    float* p = aw + (long)i * 12;
    float mx = p[0];
    #pragma unroll
    for (int j = 1; j < 12; ++j) mx = fmaxf(mx, p[j]);
    float e[12], s = 0.f;
    #pragma unroll
    for (int j = 0; j < 12; ++j) { e[j] = expf(p[j] - mx); s += e[j]; }
    float inv = 1.f / s;
    #pragma unroll
    for (int j = 0; j < 12; ++j) p[j] = e[j] * inv;
}

// ---------------------------------------------------------------------------
// Deformable sampling: one wave32 per (b,q,head); lane == channel (hd==32).
// Index math is wave-uniform; per-corner gathers of V are 128B coalesced.
// V layout: [B][S_total][256] (channel-contiguous).
// ---------------------------------------------------------------------------
__global__ __launch_bounds__(256)
void msda_sample(const float* __restrict__ V, const float* __restrict__ off,
                 const float* __restrict__ aw, const float* __restrict__ ref,
                 float* __restrict__ mid)
{
    const int bq = blockIdx.x;           // 0..BQ-1
    const int h  = threadIdx.x >> 5;     // head = wave id
    const int d  = threadIdx.x & 31;     // channel within head
    const int b  = bq / QN;

    const float rx = ref[(long)bq * 2 + 0];
    const float ry = ref[(long)bq * 2 + 1];
    const float* offp = off + ((long)bq * NH + h) * (NL * NP * 2);
    const float* awp  = aw  + ((long)bq * NH + h) * (NL * NP);
    const float* vb   = V + (long)b * STOT * DIM + h * HD + d;

    const int Wl[3]   = {80, 40, 20};
    const int Hl[3]   = {80, 40, 20};
    const int loff[3] = {0, 6400, 8000};

    float acc = 0.f;
    #pragma unroll
    for (int l = 0; l < NL; ++l) {
        const int   Wi = Wl[l], Hi = Hl[l], lo = loff[l];
        const float wf = (float)Wi, hf = (float)Hi;
        #pragma unroll
        for (int p = 0; p < NP; ++p) {
            float ox = offp[(l * NP + p) * 2 + 0];
            float oy = offp[(l * NP + p) * 2 + 1];
            float wa = awp[l * NP + p];
            // grid = (ref+off)*2-1; unnormalize (align_corners=False):
            // ix = ((gx+1)*W - 1)/2 = (ref+off)*W - 0.5
            float ix = (rx + ox) * wf - 0.5f;
            float iy = (ry + oy) * hf - 0.5f;
            float x0f = floorf(ix), y0f = floorf(iy);
            int   x0 = (int)x0f,   y0 = (int)y0f;
            float fx = ix - x0f,   fy = iy - y0f;
            float w00 = (1.f - fx) * (1.f - fy);
            float w01 = fx * (1.f - fy);
            float w10 = (1.f - fx) * fy;
            float w11 = fx * fy;
            int x1 = x0 + 1, y1 = y0 + 1;
            bool vx0 = (x0 >= 0) & (x0 < Wi), vx1 = (x1 >= 0) & (x1 < Wi);
            bool vy0 = (y0 >= 0) & (y0 < Hi), vy1 = (y1 >= 0) & (y1 < Hi);
            float s = 0.f;
            if (vx0 & vy0) s += w00 * vb[(long)(lo + y0 * Wi + x0) * DIM];
            if (vx1 & vy0) s += w01 * vb[(long)(lo + y0 * Wi + x1) * DIM];
            if (vx0 & vy1) s += w10 * vb[(long)(lo + y1 * Wi + x0) * DIM];
            if (vx1 & vy1) s += w11 * vb[(long)(lo + y1 * Wi + x1) * DIM];
            acc += wa * s;
        }
    }
    mid[(long)bq * DIM + h * HD + d] = acc;
}

static inline int ceildiv(int a, int b) { return (a + b - 1) / b; }

// ---------------------------------------------------------------------------
extern "C" void kernel_launch(void* const* d_in, const int* in_sizes, int n_in,
                              void* d_out, int out_size, void* d_ws, size_t ws_size,
                              hipStream_t stream)
{
    const float* query  = (const float*)d_in[0];
    const float* feat0  = (const float*)d_in[1];   // [4,256,80,80]
    const float* feat1  = (const float*)d_in[2];   // [4,256,40,40]
    const float* feat2  = (const float*)d_in[3];   // [4,256,20,20]
    const float* ref    = (const float*)d_in[4];
    const float* W_off  = (const float*)d_in[5];
    const float* b_off  = (const float*)d_in[6];
    const float* W_attn = (const float*)d_in[7];
    const float* b_attn = (const float*)d_in[8];
    const float* W_val  = (const float*)d_in[9];
    const float* b_val  = (const float*)d_in[10];
    const float* W_out  = (const float*)d_in[11];
    const float* b_out  = (const float*)d_in[12];
    float* out = (float*)d_out;

    // workspace carve-up (floats): offsets | attn | V | mid  (~107.5 MB)
    float* off_buf = (float*)d_ws;                         // BQ*192
    float* aw_buf  = off_buf + (size_t)BQ * 192;           // BQ*96
    float* Vbuf    = aw_buf  + (size_t)BQ * 96;            // 4*8400*256
    float* midbuf  = Vbuf    + (size_t)BATCH * STOT * DIM; // BQ*256

    dim3 blk(256);

    // sampling offsets: tanh(q@W_off + b)*0.5   M=33600 N=192 K=256
    gemm_wmma_f32<0,1><<<dim3(ceildiv(192,64), ceildiv(BQ,128), 1), blk, 0, stream>>>(
        query, W_off, b_off, off_buf, BQ, 192, 256, 256, 0, 0);
    // attention logits: q@W_attn + b            M=33600 N=96
    gemm_wmma_f32<0,0><<<dim3(ceildiv(96,64), ceildiv(BQ,128), 1), blk, 0, stream>>>(
        query, W_attn, b_attn, aw_buf, BQ, 96, 256, 256, 0, 0);
    softmax12<<<dim3((BQ * NH + 255) / 256), blk, 0, stream>>>(aw_buf, BQ * NH);

    // value projection per level; feat[b] is [S x 256] column-major (ld=S)
    gemm_wmma_f32<1,0><<<dim3(4, ceildiv(6400,128), 4), blk, 0, stream>>>(
        feat0, W_val, b_val, Vbuf,
        6400, 256, 256, 6400, (long)256 * 6400, (long)STOT * DIM);
    gemm_wmma_f32<1,0><<<dim3(4, ceildiv(1600,128), 4), blk, 0, stream>>>(
        feat1, W_val, b_val, Vbuf + 6400 * DIM,
        1600, 256, 256, 1600, (long)256 * 1600, (long)STOT * DIM);
    gemm_wmma_f32<1,0><<<dim3(4, ceildiv(400,128), 4), blk, 0, stream>>>(
        feat2, W_val, b_val, Vbuf + 8000 * DIM,
        400, 256, 256, 400, (long)256 * 400, (long)STOT * DIM);

    // bilinear sampling + attention-weighted reduction -> mid [BQ,256]
    msda_sample<<<dim3(BQ), blk, 0, stream>>>(Vbuf, off_buf, aw_buf, ref, midbuf);

    // output projection: mid @ W_out + b_out -> d_out
    gemm_wmma_f32<0,0><<<dim3(ceildiv(256,64), ceildiv(BQ,128), 1), blk, 0, stream>>>(
        midbuf, W_out, b_out, out, BQ, 256, 256, 256, 0, 0);
}